// Sparsity_48009144435553
// MI455X (gfx1250) — compile-verified
//
#include <hip/hip_runtime.h>
#include <stdint.h>

// 2:4 structured sparsity pruning (keep top-2 |x| of every 4, row-major),
// fp32 in/out. Pure streaming kernel: 512 MB traffic @ 23.3 TB/s ~= 22 us.
// Uses CDNA5 async global->LDS B128 copies (ASYNCcnt) as a double-buffered
// prefetch pipeline + non-temporal hints (512 MB stream >> 192 MB L2).

typedef __attribute__((ext_vector_type(4))) float v4f;

#define TPB 256u
#define NBLK 4096

__global__ __launch_bounds__(TPB) void prune24_kernel(
    const float* __restrict__ in,
    const int* __restrict__ um,
    const int* __restrict__ am,
    float* __restrict__ out,
    long long n_groups)
{
    __shared__ v4f buf[2][TPB];

    const bool prune = (um[0] != 0) && (am[0] != 0);

    const unsigned tid = threadIdx.x;
    const long long stride = (long long)gridDim.x * (long long)TPB;
    const long long block_base0 = (long long)blockIdx.x * (long long)TPB;

    const uint64_t gbase = (uint64_t)(uintptr_t)in;
    const uint32_t lds0 = (uint32_t)(uintptr_t)&buf[0][tid];
    const uint32_t lds1 = (uint32_t)(uintptr_t)&buf[1][tid];

    // Issue async global->LDS prefetch of chunk k for this lane's group.
    // Returns wave-uniform "this wave issued an async op for chunk k".
    auto issue = [&](long long k) -> bool {
        long long base  = block_base0 + k * stride;
        long long wbase = base + (long long)(tid & ~31u);  // wave32-uniform
        bool wave_has = (wbase < n_groups);
        if (wave_has) {
            long long g = base + (long long)tid;
            if (g < n_groups) {
                uint32_t lds  = (k & 1) ? lds1 : lds0;
                uint32_t voff = (uint32_t)(g * 16);  // byte offset, < 2^31
                // GVS mode: mem_addr = SADDR(64) + VADDR(32) ; LDS dest in %0.
                asm volatile(
                    "global_load_async_to_lds_b128 %0, %1, %2 th:TH_LOAD_NT"
                    :
                    : "v"(lds), "v"(voff), "s"(gbase)
                    : "memory");
            }
        }
        return wave_has;
    };

    bool warm = issue(0);
    (void)warm;

    for (long long k = 0; ; ++k) {
        long long base = block_base0 + k * stride;
        if (base >= n_groups) break;                 // block-uniform exit

        bool pend_next = issue(k + 1);               // wave-uniform
        if (pend_next) {
            // chunk k+1 in flight; async loads complete in order, so
            // asynccnt<=1 guarantees chunk k has landed in LDS.
            asm volatile("s_wait_asynccnt 0x1" ::: "memory");
        } else {
            asm volatile("s_wait_asynccnt 0x0" ::: "memory");
        }

        v4f x = buf[k & 1][tid];                     // ds_load_b128

        float a0 = __builtin_fabsf(x.x);
        float a1 = __builtin_fabsf(x.y);
        float a2 = __builtin_fabsf(x.z);
        float a3 = __builtin_fabsf(x.w);

        // beats(j,i): a[j]>a[i] || (a[j]==a[i] && j<i)  -> ">=" for j<i.
        int c0 = (int)(a1 >  a0) + (int)(a2 >  a0) + (int)(a3 >  a0);
        int c1 = (int)(a0 >= a1) + (int)(a2 >  a1) + (int)(a3 >  a1);
        int c2 = (int)(a0 >= a2) + (int)(a1 >= a2) + (int)(a3 >  a2);
        int c3 = (int)(a0 >= a3) + (int)(a1 >= a3) + (int)(a2 >= a3);

        v4f r;
        r.x = (prune && c0 >= 2) ? 0.0f : x.x;
        r.y = (prune && c1 >= 2) ? 0.0f : x.y;
        r.z = (prune && c2 >= 2) ? 0.0f : x.z;
        r.w = (prune && c3 >= 2) ? 0.0f : x.w;

        long long g = base + (long long)tid;
        if (g < n_groups) {
            __builtin_nontemporal_store(r, (v4f*)out + g);  // B128 NT store
        }
    }
}

extern "C" void kernel_launch(void* const* d_in, const int* in_sizes, int n_in,
                              void* d_out, int out_size, void* d_ws, size_t ws_size,
                              hipStream_t stream) {
    (void)n_in; (void)out_size; (void)d_ws; (void)ws_size;

    const float* in = (const float*)d_in[0];
    const int*   um = (const int*)d_in[1];
    const int*   am = (const int*)d_in[2];
    float*      out = (float*)d_out;

    long long n        = (long long)in_sizes[0];
    long long n_groups = n / 4;

    long long max_blocks = (n_groups + TPB - 1) / (long long)TPB;
    int blocks = NBLK;
    if ((long long)blocks > max_blocks) blocks = (int)max_blocks;
    if (blocks < 1) blocks = 1;

    prune24_kernel<<<dim3((unsigned)blocks), dim3(TPB), 0, stream>>>(
        in, um, am, out, n_groups);
}